// IrrepConditionalUnet1D_20461224198597
// MI455X (gfx1250) — compile-verified
//
#include <hip/hip_runtime.h>
#include <cmath>

// ============================================================================
// MI455X (gfx1250) implementation of the irrep-equivariant diffusion UNet.
//  - All conv1d's are implicit GEMMs on V_WMMA_F32_16X16X4_F32 (fp32 WMMA,
//    matching the fp32 reference). One wave computes a 32(M)x32(N) tile with
//    4 independent C fragments: 2 loads per WMMA, no same-D RAW chains inside
//    a K-step, K-loop unrolled x2 for latency hiding.
//  - ~60MB of weights + activations are L2-resident (192MB L2), so K-loop
//    gathers are L2 hits; 23.3TB/s HBM is touched essentially once.
//  - s2_act uses one wave32 per (b,c,t) column with __shfl_xor reductions.
// ============================================================================

typedef float v2f __attribute__((ext_vector_type(2)));
typedef float v8f __attribute__((ext_vector_type(8)));

__device__ __forceinline__ float sigmoidf_(float x){ return 1.f/(1.f+__expf(-x)); }

// ------------------------- S2 grid constants (device-built) -----------------
__global__ void k_build_Y(float* __restrict__ Y){
  int g = blockIdx.x*blockDim.x + threadIdx.x;
  if (g >= 392) return;
  int i = g / 28, j = g % 28;
  float th = 3.14159265358979323846f * (i + 0.5f) / 14.f;
  float ph = 6.28318530717958647692f * (float)j / 28.f;
  float st = __sinf(th), ct = __cosf(th);
  float x = st*__cosf(ph), y = st*__sinf(ph), z = ct;
  float* r = Y + g*9;
  r[0]=0.28209479177387814f;
  r[1]=0.4886025119029199f*y;
  r[2]=0.4886025119029199f*z;
  r[3]=0.4886025119029199f*x;
  r[4]=1.0925484305920792f*x*y;
  r[5]=1.0925484305920792f*y*z;
  r[6]=0.31539156525252005f*(3.f*z*z-1.f);
  r[7]=1.0925484305920792f*x*z;
  r[8]=0.5462742152960396f*(x*x-y*y);
}

// P = (Y^T Y)^-1 Y^T  (Y is full column rank -> equals pinv(Y))
__global__ void k_build_pinv(const float* __restrict__ Y, float* __restrict__ P){
  __shared__ float M[81];
  __shared__ float Inv[81];
  int tid = threadIdx.x;
  for (int e = tid; e < 81; e += blockDim.x){
    int r = e/9, c = e%9;
    float s = 0.f;
    for (int g=0; g<392; ++g) s += Y[g*9+r]*Y[g*9+c];
    M[e] = s;
  }
  __syncthreads();
  if (tid == 0){
    float A[9][18];
    for (int r=0;r<9;++r) for (int c=0;c<9;++c){ A[r][c]=M[r*9+c]; A[r][9+c]=(r==c)?1.f:0.f; }
    for (int col=0; col<9; ++col){
      int piv = col; float best = fabsf(A[col][col]);
      for (int r=col+1;r<9;++r){ float v=fabsf(A[r][col]); if (v>best){best=v;piv=r;} }
      if (piv != col) for (int c=0;c<18;++c){ float tmp=A[col][c]; A[col][c]=A[piv][c]; A[piv][c]=tmp; }
      float inv = 1.f/A[col][col];
      for (int c=0;c<18;++c) A[col][c]*=inv;
      for (int r=0;r<9;++r) if (r!=col){
        float f=A[r][col];
        for (int c=0;c<18;++c) A[r][c]-=f*A[col][c];
      }
    }
    for (int r=0;r<9;++r) for (int c=0;c<9;++c) Inv[r*9+c]=A[r][9+c];
  }
  __syncthreads();
  for (int idx = tid; idx < 9*392; idx += blockDim.x){
    int i = idx/392, g = idx%392;
    float s=0.f;
    for (int r=0;r<9;++r) s += Inv[i*9+r]*Y[g*9+r];
    P[i*392+g]=s;
  }
}

// ------------------------- time embedding MLP -------------------------------
__global__ void k_temb(const int* __restrict__ ts,
                       const float* __restrict__ W1, const float* __restrict__ b1,
                       const float* __restrict__ W2, const float* __restrict__ b2,
                       float* __restrict__ temb){
  __shared__ float e[128];
  __shared__ float h[512];
  int b = blockIdx.x, tid = threadIdx.x;
  float t = (float)ts[b];
  if (tid < 128){
    int i = tid & 63;
    float fr = __expf(-logf(10000.f)/63.f * (float)i);
    float a = t * fr;
    e[tid] = (tid >= 64) ? __cosf(a) : __sinf(a);
  }
  __syncthreads();
  for (int o = tid; o < 512; o += blockDim.x){
    float acc = b1[o];
    const float* w = W1 + o*128;
    for (int i=0;i<128;++i) acc += w[i]*e[i];
    float sp = (acc > 20.f) ? acc : log1pf(__expf(acc));   // softplus
    h[o] = acc * tanhf(sp);                                // mish
  }
  __syncthreads();
  for (int o = tid; o < 128; o += blockDim.x){
    float acc = b2[o];
    const float* w = W2 + o*512;
    for (int i=0;i<512;++i) acc += w[i]*h[i];
    temb[b*128 + o] = acc;
  }
}

// gf = concat([temb embedded at irrep 0, global_cond.reshape(B,128,9)], ch)
__global__ void k_gf(const float* __restrict__ temb, const float* __restrict__ gc,
                     float* __restrict__ gf){
  int idx = blockIdx.x*blockDim.x + threadIdx.x;
  if (idx >= 32*256*9) return;
  int i = idx % 9; int c = (idx/9) % 256; int b = idx/2304;
  float v;
  if (c < 128) v = (i==0) ? temb[b*128+c] : 0.f;
  else         v = gc[b*1152 + (c-128)*9 + i];
  gf[idx] = v;
}

// sample (B,T,10) -> x (B,3,9,T)
__global__ void k_assemble(const float* __restrict__ sample, float* __restrict__ X){
  int idx = blockIdx.x*blockDim.x + threadIdx.x;
  if (idx >= 32*3*9*16) return;
  int t = idx % 16; int i = (idx/16)%9; int c=(idx/144)%3; int b=idx/432;
  float v = 0.f;
  if (i>=1 && i<=3)        v = sample[(b*16+t)*10 + c*3 + (i-1)];
  else if (i==0 && c==0)   v = sample[(b*16+t)*10 + 9];
  X[idx] = v;
}

// ------------------------- WMMA implicit-GEMM irrep conv --------------------
// y[b,co,i0+j,t] = bias(l==0) + sum_{ci,dk} W_l[co,ci,dk] * X[b,ci,i0+j, map(t*stride+dk-pad)]
// Dual input pointer (x1|x2 split at C1) implements channel concat for free.
// One wave -> 32(M) x 32(N) tile; K stepped by 4 with V_WMMA_F32_16X16X4_F32.
// f32 A-frag layout (ISA 7.12.2): lanes 0-15 row=M, K={k0,k0+1}; lanes 16-31 K={k0+2,k0+3}.
template<int KW>
__global__ void k_irrep_conv_wmma(
    const float* __restrict__ x1, int C1,
    const float* __restrict__ x2, int C2,
    int Tx,
    const float* __restrict__ Wl0, const float* __restrict__ Wl1, const float* __restrict__ Wl2,
    const float* __restrict__ bias,
    float* __restrict__ y,
    int Cout, int Tout, int stride, int pad, int ups, int Bn)
{
  const int l  = blockIdx.z;
  const int m  = (l==0)?1:((l==1)?3:5);
  const int i0 = (l==0)?0:((l==1)?1:4);
  const float* __restrict__ W = (l==0)?Wl0:((l==1)?Wl1:Wl2);
  const int Cin = C1 + C2;
  const int K   = Cin*KW;
  const int N   = Bn*m*Tout;
  const int Tin = ups ? (Tx<<1) : Tx;     // effective conv input length
  const int n0  = blockIdx.y*32;
  if (n0 >= N) return;
  const int m0   = blockIdx.x*32;
  const int lane = threadIdx.x & 31;
  const int l16  = lane & 15;
  const int lh   = lane >> 4;
  const int krow = lh << 1;

  // two M-subtiles of 16 rows each
  int ar0 = m0 + l16;       if (ar0 >= Cout) ar0 = Cout-1;
  int ar1 = m0 + 16 + l16;  if (ar1 >= Cout) ar1 = Cout-1;
  const float* __restrict__ Wr0 = W + ar0*K;
  const float* __restrict__ Wr1 = W + ar1*K;

  int tc[2]; int ok[2]; int xb1[2]; int xb2[2]; int yb[2];
#pragma unroll
  for (int s=0;s<2;++s){
    int nn = n0 + (s<<4) + l16;
    ok[s] = (nn < N);
    int nc = ok[s] ? nn : (N-1);
    int t   = nc % Tout;
    int rem = nc / Tout;
    int j   = rem % m;
    int b   = rem / m;
    int ij  = i0 + j;
    tc[s]  = t;
    xb1[s] = b*C1*9*Tx + ij*Tx;
    xb2[s] = b*C2*9*Tx + ij*Tx;
    yb[s]  = b*Cout*9*Tout + ij*Tout + t;
  }

  auto bval = [&](int s, int kk) -> float {
    if (kk >= K) return 0.f;
    int ci = kk / KW;
    int dk = kk - ci*KW;
    int tin = tc[s]*stride + dk - pad;
    if (tin < 0 || tin >= Tin) return 0.f;
    int tsr = ups ? (tin >> 1) : tin;     // repeat(x,2) upsample
    if (ci < C1) return x1[xb1[s] + ci*9*Tx + tsr];
    return x2[xb2[s] + (ci-C1)*9*Tx + tsr];
  };

  v8f c00 = {0,0,0,0,0,0,0,0};   // (Msub0, Nsub0)
  v8f c01 = {0,0,0,0,0,0,0,0};   // (Msub0, Nsub1)
  v8f c10 = {0,0,0,0,0,0,0,0};   // (Msub1, Nsub0)
  v8f c11 = {0,0,0,0,0,0,0,0};   // (Msub1, Nsub1)
#pragma unroll 2
  for (int kk0 = 0; kk0 < K; kk0 += 4){
    int ka = kk0 + krow;
    bool g0 = (ka < K), g1 = (ka+1 < K);
    v2f a0; a0.x = g0 ? Wr0[ka] : 0.f; a0.y = g1 ? Wr0[ka+1] : 0.f;
    v2f a1; a1.x = g0 ? Wr1[ka] : 0.f; a1.y = g1 ? Wr1[ka+1] : 0.f;
    v2f bf0; bf0.x = bval(0, ka); bf0.y = bval(0, ka+1);
    v2f bf1; bf1.x = bval(1, ka); bf1.y = bval(1, ka+1);
    // 4 independent accumulators: no same-D RAW chain inside a K-step
    c00 = __builtin_amdgcn_wmma_f32_16x16x4_f32(false, a0, false, bf0, (short)0, c00, false, false);
    c01 = __builtin_amdgcn_wmma_f32_16x16x4_f32(false, a0, false, bf1, (short)0, c01, false, false);
    c10 = __builtin_amdgcn_wmma_f32_16x16x4_f32(false, a1, false, bf0, (short)0, c10, false, false);
    c11 = __builtin_amdgcn_wmma_f32_16x16x4_f32(false, a1, false, bf1, (short)0, c11, false, false);
  }

#pragma unroll
  for (int r=0;r<8;++r){
    int co0 = m0 + r + (lh<<3);           // C/D layout: lanes 16-31 hold M=r+8
    int co1 = co0 + 16;
    if (co0 < Cout){
      float add = (l==0) ? bias[co0] : 0.f;   // reference adds bias only to l==0
      int coff = co0*9*Tout;
      if (ok[0]) y[yb[0]+coff] = c00[r] + add;
      if (ok[1]) y[yb[1]+coff] = c01[r] + add;
    }
    if (co1 < Cout){
      float add = (l==0) ? bias[co1] : 0.f;
      int coff = co1*9*Tout;
      if (ok[0]) y[yb[0]+coff] = c10[r] + add;
      if (ok[1]) y[yb[1]+coff] = c11[r] + add;
    }
  }
}

// ------------------------- irrep RMS norm (per b,t over C x slice) ----------
__global__ void k_irrep_norm(float* __restrict__ x, int C, int T){
  __shared__ float s0[256], s1[256], s2[256];
  int bt = blockIdx.x;
  int b = bt / T, t = bt % T;
  int tid = threadIdx.x;
  float p0=0.f,p1=0.f,p2=0.f;
  int tot = C*9;
  for (int idx=tid; idx<tot; idx+=256){
    int c = idx/9, i = idx%9;
    float v = x[(b*C+c)*9*T + i*T + t];
    if (i==0) p0+=v*v; else if (i<4) p1+=v*v; else p2+=v*v;
  }
  s0[tid]=p0; s1[tid]=p1; s2[tid]=p2;
  __syncthreads();
  for (int o=128;o>0;o>>=1){
    if (tid<o){ s0[tid]+=s0[tid+o]; s1[tid]+=s1[tid+o]; s2[tid]+=s2[tid+o]; }
    __syncthreads();
  }
  float inv0 = rsqrtf(s0[0]/(float)(C  ) + 1e-5f);
  float inv1 = rsqrtf(s1[0]/(float)(C*3) + 1e-5f);
  float inv2 = rsqrtf(s2[0]/(float)(C*5) + 1e-5f);
  for (int idx=tid; idx<tot; idx+=256){
    int c = idx/9, i = idx%9;
    float inv = (i==0)?inv0:((i<4)?inv1:inv2);
    x[(b*C+c)*9*T + i*T + t] *= inv;
  }
}

// ------------------------- s2 grid activation (one wave per column) ---------
__global__ void k_s2_act(float* __restrict__ x, const float* __restrict__ Y,
                         const float* __restrict__ P, int ncols, int T){
  int wid  = (blockIdx.x*blockDim.x + threadIdx.x) >> 5;
  int lane = threadIdx.x & 31;
  if (wid >= ncols) return;
  int base = (wid / T)*9*T + (wid % T);
  float xv[9];
#pragma unroll
  for (int i=0;i<9;++i) xv[i] = x[base + i*T];
  float acc[9] = {0};
  for (int g=lane; g<392; g+=32){
    const float* yr = Y + g*9;
    float f = 0.f;
#pragma unroll
    for (int i=0;i<9;++i) f += yr[i]*xv[i];
    float s = f * sigmoidf_(f);            // silu
#pragma unroll
    for (int i=0;i<9;++i) acc[i] += P[i*392+g]*s;
  }
#pragma unroll
  for (int i=0;i<9;++i)
    for (int o=16;o>0;o>>=1) acc[i] += __shfl_xor(acc[i], o, 32);
  if (lane==0){
#pragma unroll
    for (int i=0;i<9;++i) x[base+i*T] = acc[i];
  }
}

// ------------------------- FiLM-style irrep modulation ----------------------
__global__ void k_film(float* __restrict__ out, const float* __restrict__ emb,
                       int C, int T, int total){
  int idx = blockIdx.x*blockDim.x + threadIdx.x;
  if (idx >= total) return;
  int t = idx % T; int rem = idx / T;
  int c = rem % C; int b = rem / C;
  int base = (b*C+c)*9*T + t;
  int sb = (b*2*C + 2*c)*9;                // scale ch = 2c; bias ch = 2c+1 (+9)
  float o[9], sc[9], bi[9];
#pragma unroll
  for (int i=0;i<9;++i){ o[i]=out[base+i*T]; sc[i]=emb[sb+i]; bi[i]=emb[sb+9+i]; }
  const int ss[3]={0,1,4}, ee[3]={1,4,9};
#pragma unroll
  for (int l=0;l<3;++l){
    float len=0.f, nr=0.f;
    for (int i=ss[l]; i<ee[l]; ++i){ len += o[i]*sc[i]; nr += o[i]*o[i]; }
    float inv = 1.f/(sqrtf(nr)+1e-12f);
    for (int i=ss[l]; i<ee[l]; ++i) out[base+i*T] = len*o[i]*inv + bi[i];
  }
}

__global__ void k_add(float* __restrict__ d, const float* __restrict__ a,
                      const float* __restrict__ b, int n){
  int i = blockIdx.x*blockDim.x + threadIdx.x;
  if (i < n) d[i] = a[i] + b[i];
}

// (B,3,9,16) -> (B,T,10) output
__global__ void k_final(const float* __restrict__ xf, float* __restrict__ out){
  int idx = blockIdx.x*blockDim.x + threadIdx.x;
  if (idx >= 32*16*10) return;
  int q = idx % 10; int t = (idx/10) % 16; int b = idx/160;
  float v;
  if (q < 9){
    int c = q/3, jj = q%3;
    v = xf[((b*3 + c)*9 + 1 + jj)*16 + t];
  } else {
    v = xf[((b*3 + 0)*9 + 0)*16 + t];
  }
  out[idx] = v;
}

// ============================================================================
// Host orchestration
// ============================================================================
struct ConvP { const float* W[3]; const float* b; };
struct ResP  { ConvP b0, b1, cond, res; bool has_res; };

extern "C" void kernel_launch(void* const* d_in, const int* in_sizes, int n_in,
                              void* d_out, int out_size, void* d_ws, size_t ws_size,
                              hipStream_t stream) {
  const int Bn = 32;
  const float* sample = (const float*)d_in[0];
  const int*   tstep  = (const int*)  d_in[1];
  const float* gcond  = (const float*)d_in[2];

  // ---- walk params assuming jax pytree leaf order (dict keys sorted):
  // top: down < final < mid < temb < up; conv {W,b}; res {b0,b1,cond,res};
  // down level {ds,r1,r2}; up level {r1,r2,us}; temb {W1,W2,b1,b2}.
  int pi = 3;
  auto getp   = [&](void)->const float*{ return (const float*)d_in[pi++]; };
  auto rdconv = [&](ConvP& c){ c.W[0]=getp(); c.W[1]=getp(); c.W[2]=getp(); c.b=getp(); };
  auto rdres  = [&](ResP& r, bool hr){ rdconv(r.b0); rdconv(r.b1); rdconv(r.cond);
                                       r.has_res=hr; if (hr) rdconv(r.res); };
  ConvP dsP[3]; ResP dr1[3], dr2[3];
  for (int i=0;i<3;++i){ rdconv(dsP[i]); rdres(dr1[i], true); rdres(dr2[i], false); }
  ConvP finalP; rdconv(finalP);
  ResP mid0, mid1; rdres(mid0,false); rdres(mid1,false);
  const float* tW1=getp(); const float* tW2=getp();
  const float* tb1=getp(); const float* tb2=getp();
  ResP ur1[2], ur2[2]; ConvP usP[2];
  for (int i=0;i<2;++i){ rdres(ur1[i],true); rdres(ur2[i],false); rdconv(usP[i]); }

  // ---- workspace arena (floats) ~35MB total
  float* Wb = (float*)d_ws;
  size_t off = 0;
  auto wsf = [&](size_t n){ float* p = Wb + off; off += n; return p; };
  float* Yg   = wsf(392*9);
  float* Pg   = wsf(392*9);
  float* Temb = wsf(32*128);
  float* GF   = wsf(32*256*9);
  float* E    = wsf(32*1600*9);
  const size_t BIGN = 1u<<20;
  float* B0 = wsf(BIGN); float* B1 = wsf(BIGN); float* B2 = wsf(BIGN);
  float* B3 = wsf(BIGN); float* B4 = wsf(BIGN);
  float* H1 = wsf(32*200*9*16);
  float* H2 = wsf(32*400*9*8);
  float* H3 = wsf(32*800*9*4);

  // ---- launch helpers
  auto conv = [&](const float* x1,int C1,const float* x2,int C2,int Tx,
                  const ConvP& w, float* yv, int Cout,int Tout,
                  int kw,int stride,int pad,int ups){
    dim3 g((Cout+31)/32, (Bn*5*Tout+31)/32, 3);
    if (kw==3)
      k_irrep_conv_wmma<3><<<g, dim3(32), 0, stream>>>(x1,C1,x2,C2,Tx,
          w.W[0],w.W[1],w.W[2],w.b,yv,Cout,Tout,stride,pad,ups,Bn);
    else
      k_irrep_conv_wmma<1><<<g, dim3(32), 0, stream>>>(x1,C1,x2,C2,Tx,
          w.W[0],w.W[1],w.W[2],w.b,yv,Cout,Tout,stride,pad,ups,Bn);
  };
  auto norm = [&](float* xv,int C,int T){ k_irrep_norm<<<Bn*T,256,0,stream>>>(xv,C,T); };
  auto act  = [&](float* xv,int C,int T){
    int cols=Bn*C*T; k_s2_act<<<(cols+7)/8,256,0,stream>>>(xv,Yg,Pg,cols,T); };
  auto film = [&](float* ov,const float* ev,int C,int T){
    int tot=Bn*C*T; k_film<<<(tot+255)/256,256,0,stream>>>(ov,ev,C,T,tot); };
  auto addk = [&](float* dv,const float* av,const float* bv,int n){
    k_add<<<(n+255)/256,256,0,stream>>>(dv,av,bv,n); };

  auto resblock = [&](const float* xin,int Cin,const float* xin2,int C2,int T,
                      const ResP& rp,int Cout,float* outb,
                      float* s0b,float* s1b,float* s2b){
    conv(xin,Cin,xin2,C2,T, rp.b0, s0b, Cout,T, 3,1,1,0);
    norm(s0b,Cout,T); act(s0b,Cout,T);
    conv(GF,256,nullptr,0,1, rp.cond, E, 2*Cout,1, 1,1,0,0);
    film(s0b,E,Cout,T);
    conv(s0b,Cout,nullptr,0,T, rp.b1, s1b, Cout,T, 3,1,1,0);
    norm(s1b,Cout,T); act(s1b,Cout,T);
    const float* rr = xin;
    if (rp.has_res){ conv(xin,Cin,xin2,C2,T, rp.res, s2b, Cout,T, 1,1,0,0); rr = s2b; }
    addk(outb, s1b, rr, Bn*Cout*9*T);
  };

  // ---- prologue: constants, time embedding, cond features, input assembly
  k_build_Y<<<7,64,0,stream>>>(Yg);
  k_build_pinv<<<1,128,0,stream>>>(Yg,Pg);
  k_temb<<<32,256,0,stream>>>(tstep, tW1,tb1,tW2,tb2, Temb);
  k_gf<<<(32*256*9+255)/256,256,0,stream>>>(Temb, gcond, GF);
  k_assemble<<<(32*3*9*16+255)/256,256,0,stream>>>(sample, B0);

  // ---- down path  (stride-2 SAME => pad_left 0)
  resblock(B0,  3, nullptr,0,16, dr1[0],200, B1, B2,B3,B4);
  resblock(B1,200, nullptr,0,16, dr2[0],200, H1, B2,B3,B4);
  conv(H1,200,nullptr,0,16, dsP[0], B0, 200, 8, 3,2,0,0); act(B0,200,8);

  resblock(B0,200, nullptr,0,8, dr1[1],400, B1, B2,B3,B4);
  resblock(B1,400, nullptr,0,8, dr2[1],400, H2, B2,B3,B4);
  conv(H2,400,nullptr,0,8, dsP[1], B0, 400, 4, 3,2,0,0); act(B0,400,4);

  resblock(B0,400, nullptr,0,4, dr1[2],800, B1, B2,B3,B4);
  resblock(B1,800, nullptr,0,4, dr2[2],800, H3, B2,B3,B4);
  conv(H3,800,nullptr,0,4, dsP[2], B0, 800, 4, 3,1,1,0); act(B0,800,4);   // last level: stride 1

  // ---- mid
  resblock(B0,800, nullptr,0,4, mid0,800, B1, B2,B3,B4);
  resblock(B1,800, nullptr,0,4, mid1,800, B0, B2,B3,B4);

  // ---- up path (channel concat via dual-input conv)
  resblock(B0,800, H3,800,4, ur1[0],400, B1, B2,B3,B4);
  resblock(B1,400, nullptr,0,4, ur2[0],400, B0, B2,B3,B4);
  conv(B0,400,nullptr,0,4, usP[0], B1, 400, 8, 3,1,1,1); act(B1,400,8);   // upsample x2

  resblock(B1,400, H2,400,8, ur1[1],200, B0, B2,B3,B4);
  resblock(B0,200, nullptr,0,8, ur2[1],200, B1, B2,B3,B4);
  conv(B1,200,nullptr,0,8, usP[1], B0, 200, 16, 3,1,1,1); act(B0,200,16);

  // ---- final conv + output assembly
  conv(B0,200,nullptr,0,16, finalP, B1, 3, 16, 3,1,1,0);
  k_final<<<(32*16*10+255)/256,256,0,stream>>>(B1, (float*)d_out);
}